// Graph_ConvNet_LeNet5_40819369181301
// MI455X (gfx1250) — compile-verified
//
#include <hip/hip_runtime.h>
#include <cstdint>
#include <cstddef>

// ---------------------------------------------------------------------------
// Graph ChebNet (LeNet5-style) forward for MI455X / gfx1250, fp32 + WMMA f32.
//   D=16384 vertices, B=64, K=25, F1=32, F2=64, FC1=512, FC2=10
// ---------------------------------------------------------------------------

typedef __attribute__((ext_vector_type(2))) float v2f;
typedef __attribute__((ext_vector_type(4))) float v4f;
typedef __attribute__((ext_vector_type(8))) float v8f;

#define D_V    16384
#define BATCH  64
#define KCHEB  25
#define F1     32
#define V2C    4096
#define F2     64
#define V3C    1024
#define ROW2   2048           // layer-2 activation row width = B*F1 (b*32+fin)
#define FC1IN  65536          // V3C * F2
#define FC1F   512
#define FC2F   10
#define KSPLIT 64             // FC1 K-split factor

// -------------------- CSR row-pointer build (rows are sorted) --------------
__global__ void build_rowptr(const int* __restrict__ rows, int nnz, int V,
                             int* __restrict__ rowptr) {
  int r = blockIdx.x * blockDim.x + threadIdx.x;
  if (r > V) return;
  int lo = 0, hi = nnz;
  while (lo < hi) { int mid = (lo + hi) >> 1; if (rows[mid] < r) lo = mid + 1; else hi = mid; }
  rowptr[r] = lo;
}

// -------------------- x (B,D) -> x0 (D,B)  (cheb1[0]) ----------------------
__global__ void transpose_x(const float* __restrict__ x, float* __restrict__ x0) {
  int idx = blockIdx.x * 256 + threadIdx.x;          // v*64 + b, 1,048,576 total
  int b = idx & 63, v = idx >> 6;
  x0[idx] = x[(size_t)b * D_V + v];
}

// -------------------- layer-1 SpMM Chebyshev step (b128 per edge) ----------
// T_k = (first ? Ahat(prev) : 2*Ahat(prev) - prevprev), Ahat(y) = agg(y) - y
// Block: 64 thr = 4 vertices x 16 float4-columns of the 64-wide row.
__global__ __launch_bounds__(64)
void spmm_cheb1(const int* __restrict__ rowptr, const int* __restrict__ cols,
                const float* __restrict__ vals, const float* __restrict__ prev,
                const float* __restrict__ prevprev, float* __restrict__ outT, int first) {
  const int tid = threadIdx.x;
  const int c4 = tid & 15, vi = tid >> 4;
  const int v = blockIdx.x * 4 + vi;
  const int e0 = rowptr[v], e1 = rowptr[v + 1];
  const v4f* prev4 = (const v4f*)prev;
  v4f agg = {0.f, 0.f, 0.f, 0.f};
  for (int e = e0; e < e1; ++e) {
    float s = vals[e];
    v4f sv = {s, s, s, s};
    agg += sv * prev4[(size_t)cols[e] * 16 + c4];
  }
  v4f hat = agg - prev4[(size_t)v * 16 + c4];
  v4f res;
  if (first) res = hat;
  else {
    v4f two = {2.f, 2.f, 2.f, 2.f};
    res = two * hat - ((const v4f*)prevprev)[(size_t)v * 16 + c4];
  }
  ((v4f*)outT)[(size_t)v * 16 + c4] = res;
}

// -------------------- layer-1 basis GEMM + bias + relu + maxpool(4) --------
// pooled1[v2][b*32+fin] = max_{i<4} relu( b1[fin] + sum_k cheb1[k][4v2+i][b]*W1[fin][k] )
__global__ __launch_bounds__(64)
void cheb1_fc_pool(const float* __restrict__ cheb1, const float* __restrict__ W1,
                   const float* __restrict__ b1, float* __restrict__ pooled1) {
  __shared__ float sW[F1 * KCHEB];
  __shared__ float sB[F1];
  const int b = threadIdx.x, v2 = blockIdx.x;
  for (int i = b; i < F1 * KCHEB; i += 64) sW[i] = W1[i];
  if (b < F1) sB[b] = b1[b];
  __syncthreads();

  float best[F1];
#pragma unroll
  for (int f = 0; f < F1; ++f) best[f] = 0.0f;        // relu floor
  for (int i = 0; i < 4; ++i) {
    const int v = v2 * 4 + i;
    float xv[KCHEB];
#pragma unroll
    for (int k = 0; k < KCHEB; ++k)
      xv[k] = cheb1[((size_t)k * D_V + v) * BATCH + b];
    for (int f = 0; f < F1; ++f) {
      float acc = sB[f];
#pragma unroll
      for (int k = 0; k < KCHEB; ++k) acc = fmaf(xv[k], sW[f * KCHEB + k], acc);
      best[f] = fmaxf(best[f], acc);
    }
  }
  float* po = pooled1 + (size_t)v2 * ROW2 + b * F1;   // (v, b*32+fin)
#pragma unroll
  for (int f = 0; f < F1; ++f) po[f] = best[f];
}

// -------------------- W2 (64, fin*25+k) -> W2t (k, f, fin) -----------------
__global__ void prep_w2t(const float* __restrict__ W2, float* __restrict__ W2t) {
  int idx = blockIdx.x * 256 + threadIdx.x;           // 25*64*32 = 51200
  if (idx >= KCHEB * F1 * F2) return;
  int k = idx >> 11, rem = idx & 2047, f = rem >> 5, fin = rem & 31;
  W2t[idx] = W2[(size_t)f * (F1 * KCHEB) + fin * KCHEB + k];
}

// ---------------------------------------------------------------------------
// out2 lives in a WMMA fragment-major layout so accumulator RMW is b128:
//   element (r = b*4096+v, f):  tr=r>>4, m=r&15, nt=f>>4, n=f&15
//   lane L = n + 16*(m>>3), vgpr j = m&7
//   idx = (tr*4 + nt)*256 + L*8 + j
// ---------------------------------------------------------------------------
__global__ void init_out2(const float* __restrict__ b2, float* __restrict__ out2F) {
  size_t idx = (size_t)blockIdx.x * 256 + threadIdx.x; // 16,777,216
  int L = (int)((idx >> 3) & 31), nt = (int)((idx >> 8) & 3);
  out2F[idx] = b2[nt * 16 + (L & 15)];
}

// -------------------- layer-2 SpMM Chebyshev step (b128 per edge) ----------
__global__ __launch_bounds__(128)
void spmm_cheb2(const int* __restrict__ rowptr, const int* __restrict__ cols,
                const float* __restrict__ vals, const float* __restrict__ prev,
                const float* __restrict__ prevprev, float* __restrict__ outT, int first) {
  const int v = blockIdx.x;                            // 0..4095
  const int c4 = blockIdx.y * 128 + threadIdx.x;       // 0..511 (float4 cols)
  const int e0 = rowptr[v], e1 = rowptr[v + 1];
  const v4f* prev4 = (const v4f*)prev;
  v4f agg = {0.f, 0.f, 0.f, 0.f};
  for (int e = e0; e < e1; ++e) {
    float s = vals[e];
    v4f sv = {s, s, s, s};
    agg += sv * prev4[(size_t)cols[e] * (ROW2 / 4) + c4];
  }
  v4f hat = agg - prev4[(size_t)v * (ROW2 / 4) + c4];
  v4f res;
  if (first) res = hat;
  else {
    v4f two = {2.f, 2.f, 2.f, 2.f};
    res = two * hat - ((const v4f*)prevprev)[(size_t)v * (ROW2 / 4) + c4];
  }
  ((v4f*)outT)[(size_t)v * (ROW2 / 4) + c4] = res;
}

// -------------------- layer-2 accumulating WMMA GEMM -----------------------
// out2[(b*4096+v)][f] += sum_fin Tk[v][b*32+fin] * W2[f][fin*25+k]
// Block: 128 thr (4 waves). Each wave: 32 rows x 64 cols (2 m-tiles, 4 n-tiles).
// LDS stride 36: b128-aligned fills, b64-aligned conflict-free fragment reads.
#define LDS_STRIDE 36
__global__ __launch_bounds__(128)
void gemm2_acc(const float* __restrict__ Tk, const float* __restrict__ W2t,
               float* __restrict__ out2F, int k) {
  __shared__ float As[128 * LDS_STRIDE];   // [row][fin]
  __shared__ float Bs[F2 * LDS_STRIDE];    // [f][fin]  (transposed: frag = b64)
  const int tid = threadIdx.x;
  const int w = tid >> 5, lane = tid & 31;
  const int n = lane & 15, hi = lane >> 4;       // n doubles as A-row-in-tile (m)
  const int r0 = blockIdx.x * 128;               // rows r = b*4096 + v, same b
  const int b = r0 >> 12, v0 = r0 & 4095;

  { // A tile 128x32 and B tile 64x32, both filled with b128 loads/stores
    const int fin4 = (tid & 7) * 4, rb = tid >> 3;
#pragma unroll
    for (int i = 0; i < 8; ++i) {
      int row = rb + 16 * i;
      *(v4f*)&As[row * LDS_STRIDE + fin4] =
          *(const v4f*)&Tk[(size_t)(v0 + row) * ROW2 + b * F1 + fin4];
    }
#pragma unroll
    for (int i = 0; i < 4; ++i) {
      int f = rb + 16 * i;
      *(v4f*)&Bs[f * LDS_STRIDE + fin4] =
          *(const v4f*)&W2t[(size_t)k * (F1 * F2) + f * F1 + fin4];
    }
  }
  __syncthreads();

  v8f acc[2][4];
  const size_t tr0 = (size_t)(r0 >> 4) + w * 2;
#pragma unroll
  for (int mt = 0; mt < 2; ++mt)                 // load C: 2x b128 per tile
#pragma unroll
    for (int nt = 0; nt < 4; ++nt) {
      const float* p = out2F + (((tr0 + mt) * 4 + nt) << 8) + lane * 8;
#pragma unroll
      for (int j = 0; j < 8; ++j) acc[mt][nt][j] = p[j];
    }

  for (int kk = 0; kk < F1; kk += 4) {
    v2f a0 = *(const v2f*)&As[(w * 32 + n) * LDS_STRIDE + kk + 2 * hi];
    v2f a1 = *(const v2f*)&As[(w * 32 + 16 + n) * LDS_STRIDE + kk + 2 * hi];
#pragma unroll
    for (int nt = 0; nt < 4; ++nt) {
      v2f bf = *(const v2f*)&Bs[(nt * 16 + n) * LDS_STRIDE + kk + 2 * hi];
      acc[0][nt] = __builtin_amdgcn_wmma_f32_16x16x4_f32(
          false, a0, false, bf, (short)0, acc[0][nt], false, false);
      acc[1][nt] = __builtin_amdgcn_wmma_f32_16x16x4_f32(
          false, a1, false, bf, (short)0, acc[1][nt], false, false);
    }
  }

#pragma unroll
  for (int mt = 0; mt < 2; ++mt)
#pragma unroll
    for (int nt = 0; nt < 4; ++nt) {
      float* p = out2F + (((tr0 + mt) * 4 + nt) << 8) + lane * 8;
#pragma unroll
      for (int j = 0; j < 8; ++j) p[j] = acc[mt][nt][j];
    }
}

// -------------------- relu + maxpool(4) -> pooled2 (B, V3*64+f) ------------
// The 4 pool candidates share a fragment octet -> single b128 load.
__global__ void relu_pool2(const float* __restrict__ out2F, float* __restrict__ pooled2) {
  int idx = blockIdx.x * 256 + threadIdx.x;            // b*65536 + v3*64 + f
  int f = idx & 63, v3 = (idx >> 6) & 1023, b = idx >> 16;
  int nt = f >> 4, n = f & 15;
  size_t tr = (size_t)b * 256 + (v3 >> 2);
  int m0 = (v3 & 3) * 4;                               // rows m0..m0+3, same octet
  int L = n + ((m0 >> 3) << 4), j0 = m0 & 7;
  v4f q = *(const v4f*)&out2F[((tr * 4 + nt) << 8) + (size_t)L * 8 + j0];
  float mx = fmaxf(fmaxf(q.x, q.y), fmaxf(q.z, q.w));
  pooled2[idx] = fmaxf(mx, 0.0f);
}

// -------------------- FC1 WMMA with K-split --------------------------------
// partial[z][b][n] = sum_{c in chunk z} pooled2[b][c] * fc1_W[n][c]
__global__ __launch_bounds__(32)
void fc1_wmma(const float* __restrict__ H, const float* __restrict__ W,
              float* __restrict__ partial) {
  const int lane = threadIdx.x;
  const int m = lane & 15, hi = lane >> 4;
  const int mt = blockIdx.x;             // 0..3   (M=64)
  const int nt = blockIdx.y;             // 0..31  (N=512)
  const int z  = blockIdx.z;             // 0..63  (K chunks of 1024)
  v8f acc = {};
  const float* pa = H + (size_t)(mt * 16 + m) * FC1IN + (size_t)z * 1024 + 2 * hi;
  const float* pb = W + (size_t)(nt * 16 + m) * FC1IN + (size_t)z * 1024 + 2 * hi;
  for (int c = 0; c < 1024; c += 4) {
    v2f a = *(const v2f*)(pa + c);
    v2f b = *(const v2f*)(pb + c);
    acc = __builtin_amdgcn_wmma_f32_16x16x4_f32(
        false, a, false, b, (short)0, acc, false, false);
  }
  float* pp = partial + (size_t)z * (64 * FC1F) + (size_t)(mt * 16 + 8 * hi) * FC1F
            + nt * 16 + m;
#pragma unroll
  for (int j = 0; j < 8; ++j) pp[(size_t)j * FC1F] = acc[j];
}

__global__ void fc1_reduce(const float* __restrict__ partial, const float* __restrict__ bias,
                           float* __restrict__ h1) {
  int idx = blockIdx.x * 256 + threadIdx.x;            // b*512 + n (32768)
  float acc = bias[idx & (FC1F - 1)];
  for (int zz = 0; zz < KSPLIT; ++zz) acc += partial[(size_t)zz * (64 * FC1F) + idx];
  h1[idx] = fmaxf(acc, 0.0f);
}

// -------------------- FC2 --------------------------------------------------
__global__ void fc2_kernel(const float* __restrict__ h1, const float* __restrict__ W,
                           const float* __restrict__ bias, float* __restrict__ out) {
  int b = blockIdx.x, j = threadIdx.x;
  if (j < FC2F) {
    float acc = bias[j];
    const float* hp = h1 + (size_t)b * FC1F;
    const float* wp = W + (size_t)j * FC1F;
    for (int n = 0; n < FC1F; ++n) acc = fmaf(hp[n], wp[n], acc);
    out[b * FC2F + j] = acc;
  }
}

// ---------------------------------------------------------------------------
extern "C" void kernel_launch(void* const* d_in, const int* in_sizes, int n_in,
                              void* d_out, int out_size, void* d_ws, size_t ws_size,
                              hipStream_t stream) {
  const float* x       = (const float*)d_in[0];
  const int*   L0_rows = (const int*)  d_in[1];
  const int*   L0_cols = (const int*)  d_in[2];
  const float* L0_vals = (const float*)d_in[3];
  const int*   L2_rows = (const int*)  d_in[4];
  const int*   L2_cols = (const int*)  d_in[5];
  const float* L2_vals = (const float*)d_in[6];
  const float* cl1_W   = (const float*)d_in[7];
  const float* cl1_b   = (const float*)d_in[8];
  const float* cl2_W   = (const float*)d_in[9];
  const float* cl2_b   = (const float*)d_in[10];
  const float* fc1_W   = (const float*)d_in[11];
  const float* fc1_b   = (const float*)d_in[12];
  const float* fc2_W   = (const float*)d_in[13];
  const float* fc2_b   = (const float*)d_in[14];
  const int nnz0 = in_sizes[1], nnz2 = in_sizes[4];

  // ---- workspace carve-out (256B aligned), ~332 MB total ----
  char* base = (char*)d_ws;
  size_t off = 0;
  auto carve = [&](size_t bytes) {
    char* p = base + off;
    off += (bytes + 255) & ~(size_t)255;
    return p;
  };
  int*   rowptr0 = (int*)  carve((size_t)(D_V + 1) * 4);
  int*   rowptr2 = (int*)  carve((size_t)(V2C + 1) * 4);
  float* cheb1   = (float*)carve((size_t)KCHEB * D_V * BATCH * 4);      // 100 MB
  float* pooled1 = (float*)carve((size_t)V2C * ROW2 * 4);               // 32 MB
  float* tb0     = (float*)carve((size_t)V2C * ROW2 * 4);
  float* tb1     = (float*)carve((size_t)V2C * ROW2 * 4);
  float* tb2     = (float*)carve((size_t)V2C * ROW2 * 4);
  float* w2t     = (float*)carve((size_t)KCHEB * F1 * F2 * 4);
  float* out2    = (float*)carve((size_t)BATCH * V2C * F2 * 4);         // 64 MB
  float* pooled2 = (float*)carve((size_t)BATCH * FC1IN * 4);            // 16 MB
  float* partial = (float*)carve((size_t)KSPLIT * 64 * FC1F * 4);       // 8 MB
  float* h1      = (float*)carve((size_t)BATCH * FC1F * 4);
  (void)ws_size; (void)n_in; (void)out_size;

  const size_t CH1 = (size_t)D_V * BATCH;   // elements per layer-1 basis

  // ---- graph CSR prep ----
  build_rowptr<<<(D_V + 1 + 255) / 256, 256, 0, stream>>>(L0_rows, nnz0, D_V, rowptr0);
  build_rowptr<<<(V2C + 1 + 255) / 256, 256, 0, stream>>>(L2_rows, nnz2, V2C, rowptr2);

  // ---- layer 1: Chebyshev recurrence (materialize all 25 bases) ----
  transpose_x<<<(D_V * BATCH) / 256, 256, 0, stream>>>(x, cheb1);       // T0
  for (int k = 1; k < KCHEB; ++k) {
    const float* prev = cheb1 + (size_t)(k - 1) * CH1;
    const float* pp   = (k >= 2) ? cheb1 + (size_t)(k - 2) * CH1 : cheb1;
    spmm_cheb1<<<D_V / 4, 64, 0, stream>>>(rowptr0, L0_cols, L0_vals, prev, pp,
                                           cheb1 + (size_t)k * CH1, k == 1);
  }
  // basis GEMM + bias + relu + maxpool(4)
  cheb1_fc_pool<<<V2C, 64, 0, stream>>>(cheb1, cl1_W, cl1_b, pooled1);

  // ---- layer 2: rolling Chebyshev + accumulating WMMA GEMMs ----
  prep_w2t<<<(KCHEB * F1 * F2 + 255) / 256, 256, 0, stream>>>(cl2_W, w2t);
  init_out2<<<(BATCH * V2C * F2) / 256, 256, 0, stream>>>(cl2_b, out2);

  gemm2_acc<<<(BATCH * V2C) / 128, 128, 0, stream>>>(pooled1, w2t, out2, 0); // k=0

  spmm_cheb2<<<dim3(V2C, 4), 128, 0, stream>>>(rowptr2, L2_cols, L2_vals,
                                               pooled1, pooled1, tb0, 1);
  gemm2_acc<<<(BATCH * V2C) / 128, 128, 0, stream>>>(tb0, w2t, out2, 1);     // k=1

  float* tbuf[3] = {tb0, tb1, tb2};
  const float* ppv = pooled1;   // T_{k-2}
  const float* pv  = tb0;       // T_{k-1}
  for (int k = 2; k < KCHEB; ++k) {
    float* nx = tbuf[(k - 1) % 3];
    spmm_cheb2<<<dim3(V2C, 4), 128, 0, stream>>>(rowptr2, L2_cols, L2_vals,
                                                 pv, ppv, nx, 0);
    gemm2_acc<<<(BATCH * V2C) / 128, 128, 0, stream>>>(nx, w2t, out2, k);
    ppv = pv; pv = nx;
  }

  relu_pool2<<<(BATCH * FC1IN) / 256, 256, 0, stream>>>(out2, pooled2);

  // ---- FC1 (WMMA, K-split) + FC2 ----
  fc1_wmma<<<dim3(4, FC1F / 16, KSPLIT), 32, 0, stream>>>(pooled2, fc1_W, partial);
  fc1_reduce<<<(BATCH * FC1F) / 256, 256, 0, stream>>>(partial, fc1_b, h1);
  fc2_kernel<<<BATCH, 32, 0, stream>>>(h1, fc2_W, fc2_b, (float*)d_out);
}